// TemporalGraphTransformer_20469814133176
// MI455X (gfx1250) — compile-verified
//
#include <hip/hip_runtime.h>
#include <hip/hip_bf16.h>
#include <math.h>

typedef __attribute__((ext_vector_type(2))) float v2f;
typedef __attribute__((ext_vector_type(8))) float v8f;

#define D256 256

// ---------------------------------------------------------------------------
// 256x256 transpose + k-pair interleave pack:
//   Wtp[(k/2)*512 + 2*n + (k&1)] = W[n][k]
// so a WMMA B-fragment {B[k][n], B[k+1][n]} is one aligned 8-byte load.
// ---------------------------------------------------------------------------
__global__ __launch_bounds__(256) void k_transpose_pack256(const float* __restrict__ W,
                                                           float* __restrict__ Wtp) {
    __shared__ float tile[32][33];
    const int tx = threadIdx.x;      // 0..31
    const int ty = threadIdx.y;      // 0..7
    const int bx = blockIdx.x;       // col tile of W
    const int by = blockIdx.y;       // row tile of W
    int x = bx * 32 + tx;
    #pragma unroll
    for (int j = 0; j < 4; ++j) {
        int y = by * 32 + ty + 8 * j;
        tile[ty + 8 * j][tx] = W[y * D256 + x];
    }
    __syncthreads();
    #pragma unroll
    for (int j = 0; j < 4; ++j) {
        int k = bx * 32 + ty + 8 * j;   // row of Wt (= col of W)
        int n = by * 32 + tx;           // col of Wt (= row of W)
        Wtp[(k >> 1) * 512 + 2 * n + (k & 1)] = tile[tx][ty + 8 * j];
    }
}

// ---------------------------------------------------------------------------
// WMMA fp32 GEMM: Y[m][n] = sum_k X[m][k] * W[n][k] (+ bias[n]),
// weights pre-packed as Wtp (k-pair interleaved transpose).
// Block = 256 threads = 8 waves; block covers 128 rows; wave w covers rows
// [base+16w, base+16w+16) x all 256 cols (16 accumulator tiles, 128 VGPRs).
// Weight K-slices are staged into double-buffered LDS with
// GLOBAL_LOAD_ASYNC_TO_LDS_B128 (ASYNCcnt), overlapping the WMMA work.
// A/B fragments are software-pipelined one k-step ahead so the 16 DS loads +
// 1 global load of step k0+4 issue before the 16 WMMAs of step k0.
// ---------------------------------------------------------------------------
__global__ __launch_bounds__(256) void k_gemm_xwt(const float* __restrict__ X,
                                                  const float* __restrict__ Wtp,
                                                  const float* __restrict__ bias,
                                                  float* __restrict__ Y,
                                                  int nrows) {
    extern __shared__ float sB[];               // 2 x 16384 floats (128 KB)
    const int t    = threadIdx.x;
    const int wave = t >> 5;
    const int lane = t & 31;
    const int hi   = lane >> 4;                 // half-wave select (K offset +2)
    const int ln   = lane & 15;
    const int rowBase = blockIdx.x * 128 + wave * 16;

    v8f acc[16];
    #pragma unroll
    for (int i = 0; i < 16; ++i) acc[i] = (v8f){0.f,0.f,0.f,0.f,0.f,0.f,0.f,0.f};

    // A-matrix 16x4 fp32 layout: lanes 0-15 -> M=ln, K=k0..k0+1;
    // lanes 16-31 -> M=ln, K=k0+2..k0+3.
    int arow = rowBase + ln;
    if (arow >= nrows) arow = (nrows > 0) ? (nrows - 1) : 0;  // clamp (stores guarded)
    const float* Xrow = X + (size_t)arow * D256;

    const unsigned ldsBase = (unsigned)(unsigned long long)(void*)&sB[0];
    const unsigned chunk   = (unsigned)t * 16u;   // 16B per thread per round

    // Issue one 64KB slice stage: 16 async b128 per thread (4KB stride rounds).
    auto issue_slice = [&](int ks, int buf) {
        const char* g = (const char*)(Wtp + (size_t)ks * 16384);
        unsigned lds = ldsBase + (unsigned)buf * 65536u + chunk;
        #pragma unroll
        for (int j = 0; j < 16; ++j) {
            unsigned long long ga = (unsigned long long)(g + chunk + j * 4096);
            unsigned la = lds + (unsigned)j * 4096u;
            asm volatile("global_load_async_to_lds_b128 %0, %1, off"
                         :: "v"(la), "v"(ga) : "memory");
        }
    };

    issue_slice(0, 0);

    for (int ks = 0; ks < 4; ++ks) {
        const int buf = ks & 1;
        if (ks < 3) issue_slice(ks + 1, buf ^ 1);

        // Wait for this slice's staging (next slice's 16 may stay in flight).
        if (ks < 3) asm volatile("s_wait_asynccnt 0x10" ::: "memory");
        else        asm volatile("s_wait_asynccnt 0x0"  ::: "memory");
        __syncthreads();

        const float* sBcur = sB + buf * 16384;
        const float* Arow  = Xrow + ks * 64 + 2 * hi;
        const float* Bbase = sBcur + hi * 512 + 2 * ln;

        // Prologue: fragments for k0 = 0.
        v2f aCur = __builtin_nontemporal_load((const v2f*)(Arow));
        v2f bCur[16];
        #pragma unroll
        for (int nt = 0; nt < 16; ++nt)
            bCur[nt] = ((const v2f*)Bbase)[nt * 16];

        #pragma unroll
        for (int k0 = 0; k0 < 64; k0 += 4) {
            v2f aNxt;
            v2f bNxt[16];
            if (k0 < 60) {                      // prefetch k-step k0+4
                aNxt = __builtin_nontemporal_load((const v2f*)(Arow + k0 + 4));
                const v2f* bp = (const v2f*)(Bbase + ((k0 + 4) >> 1) * 512);
                #pragma unroll
                for (int nt = 0; nt < 16; ++nt) bNxt[nt] = bp[nt * 16];
            }
            #pragma unroll
            for (int nt = 0; nt < 16; ++nt)
                acc[nt] = __builtin_amdgcn_wmma_f32_16x16x4_f32(
                    false, aCur, false, bCur[nt], (short)0, acc[nt], false, false);
            if (k0 < 60) {
                aCur = aNxt;
                #pragma unroll
                for (int nt = 0; nt < 16; ++nt) bCur[nt] = bNxt[nt];
            }
        }
        __syncthreads();   // all waves done with buf before it is re-staged
    }

    // C/D layout: VGPR r, lanes 0-15 -> M=r, lanes 16-31 -> M=r+8; N = lane&15.
    #pragma unroll
    for (int nt = 0; nt < 16; ++nt) {
        const int col = nt * 16 + ln;
        const float bb = bias ? bias[col] : 0.f;
        #pragma unroll
        for (int r = 0; r < 8; ++r) {
            const int row = rowBase + r + 8 * hi;
            if (row < nrows)
                __builtin_nontemporal_store(acc[nt][r] + bb,
                                            &Y[(size_t)row * D256 + col]);
        }
    }
}

// ---------------------------------------------------------------------------
// Prefix scan of edge_len -> segment offsets (single block)
// ---------------------------------------------------------------------------
__global__ __launch_bounds__(256) void k_scan_offsets(const int* __restrict__ el,
                                                      int* __restrict__ off, int S) {
    __shared__ int chunk[256];
    const int t = threadIdx.x;
    const int per = (S + 255) / 256;
    int s = 0;
    for (int i = 0; i < per; ++i) {
        int idx = t * per + i;
        if (idx < S) s += el[idx];
    }
    chunk[t] = s;
    __syncthreads();
    if (t == 0) {
        int run = 0;
        for (int i = 0; i < 256; ++i) { int c = chunk[i]; chunk[i] = run; run += c; }
        off[S] = run;
    }
    __syncthreads();
    int base = chunk[t];
    for (int i = 0; i < per; ++i) {
        int idx = t * per + i;
        if (idx < S) { off[idx] = base; base += el[idx]; }
    }
}

// ---------------------------------------------------------------------------
// Per-edge logits: logits[e] = dot(qp[seg[e]], kp[e]) * scale  (wave per edge)
// ---------------------------------------------------------------------------
__global__ __launch_bounds__(256) void k_edge_logits(const float* __restrict__ qp,
                                                     const float* __restrict__ kp,
                                                     const int* __restrict__ seg,
                                                     float* __restrict__ logits,
                                                     int E, float scale) {
    const int e = blockIdx.x * 8 + (threadIdx.x >> 5);
    const int lane = threadIdx.x & 31;
    if (e >= E) return;
    const float* a = qp + (size_t)seg[e] * D256;
    const float* b = kp + (size_t)e * D256;
    float acc = 0.f;
    #pragma unroll
    for (int j = 0; j < 8; ++j) acc += a[lane + 32 * j] * b[lane + 32 * j];
    #pragma unroll
    for (int m = 16; m >= 1; m >>= 1) acc += __shfl_xor(acc, m, 32);
    if (lane == 0) logits[e] = acc * scale;
}

// ---------------------------------------------------------------------------
// Segment softmax + weighted sum of vp. One block (256 threads) per node.
// ---------------------------------------------------------------------------
__global__ __launch_bounds__(256) void k_node_attn(const float* __restrict__ logits,
                                                   const float* __restrict__ vp,
                                                   const int* __restrict__ off,
                                                   float* __restrict__ node_vals) {
    __shared__ float red[256];
    __shared__ float attn[512];
    const int s = blockIdx.x;
    const int t = threadIdx.x;
    const int beg = off[s], end = off[s + 1];
    const int n = end - beg;

    float m = -INFINITY;
    for (int e = beg + t; e < end; e += 256) m = fmaxf(m, logits[e]);
    red[t] = m;
    __syncthreads();
    for (int st = 128; st > 0; st >>= 1) {
        if (t < st) red[t] = fmaxf(red[t], red[t + st]);
        __syncthreads();
    }
    m = red[0];
    __syncthreads();

    float d = 0.f;
    for (int e = beg + t; e < end; e += 256) d += expf(logits[e] - m);
    red[t] = d;
    __syncthreads();
    for (int st = 128; st > 0; st >>= 1) {
        if (t < st) red[t] += red[t + st];
        __syncthreads();
    }
    d = red[0];
    __syncthreads();
    const float inv = (n > 0) ? (1.f / d) : 0.f;

    for (int e = beg + t; e < end; e += 256) {
        int i = e - beg;
        if (i < 512) attn[i] = expf(logits[e] - m) * inv;
    }
    __syncthreads();

    float acc = 0.f;
    for (int e = beg; e < end; ++e) {
        int i = e - beg;
        float w = (i < 512) ? attn[i] : (expf(logits[e] - m) * inv);
        acc += w * vp[(size_t)e * D256 + t];
    }
    node_vals[(size_t)s * D256 + t] = acc;
}

// ---------------------------------------------------------------------------
// Path attention (all tiny, single blocks)
// ---------------------------------------------------------------------------
__global__ __launch_bounds__(256) void k_pair_q(const float* __restrict__ q,
                                                const float* __restrict__ W_qpair,
                                                const int* __restrict__ pair,
                                                float* __restrict__ pair_q) {
    const int j = threadIdx.x;
    const int src = pair[1], tar = pair[0];
    const float* Wr = W_qpair + (size_t)j * 512;
    float acc = 0.f;
    for (int i = 0; i < 256; ++i) acc += Wr[i] * q[(size_t)src * D256 + i];
    for (int i = 0; i < 256; ++i) acc += Wr[256 + i] * q[(size_t)tar * D256 + i];
    pair_q[j] = acc;
}

__global__ __launch_bounds__(256) void k_path_softmax(const float* __restrict__ pk,
                                                      const float* __restrict__ pair_q,
                                                      float* __restrict__ p_attn,
                                                      int P, float scale) {
    __shared__ float sl[256];
    const int t = threadIdx.x;
    if (t < P) {
        float a = 0.f;
        for (int i = 0; i < 256; ++i) a += pk[(size_t)t * D256 + i] * pair_q[i];
        sl[t] = a * scale;
    }
    __syncthreads();
    if (t == 0) {
        float m = -INFINITY;
        for (int p = 0; p < P; ++p) m = fmaxf(m, sl[p]);
        float s = 0.f;
        for (int p = 0; p < P; ++p) { sl[p] = expf(sl[p] - m); s += sl[p]; }
        float inv = 1.f / s;
        for (int p = 0; p < P; ++p) sl[p] *= inv;
    }
    __syncthreads();
    if (t < P) p_attn[t] = sl[t];
}

__global__ __launch_bounds__(256) void k_path_feat(const float* __restrict__ p_attn,
                                                   const float* __restrict__ pv,
                                                   const float* __restrict__ W_out,
                                                   const float* __restrict__ b_out,
                                                   float* __restrict__ path_feat, int P) {
    __shared__ float pf[256];
    const int t = threadIdx.x;
    float a = 0.f;
    for (int p = 0; p < P; ++p) a += p_attn[p] * pv[(size_t)p * D256 + t];
    pf[t] = a;
    __syncthreads();
    float acc = b_out[t];
    const float* Wr = W_out + (size_t)t * D256;
    for (int i = 0; i < 256; ++i) acc += pf[i] * Wr[i];
    path_feat[t] = acc;
}

// ---------------------------------------------------------------------------
// Output assembly
// ---------------------------------------------------------------------------
__global__ __launch_bounds__(256) void k_copy_q(const float* __restrict__ q,
                                                float* __restrict__ out, int n4) {
    int i = blockIdx.x * 256 + threadIdx.x;
    if (i < n4) ((float4*)out)[i] = ((const float4*)q)[i];
}

__global__ __launch_bounds__(256) void k_combine(const float* __restrict__ node_out,
                                                 const float* __restrict__ q,
                                                 const int* __restrict__ token_ids,
                                                 const int* __restrict__ edge_len,
                                                 float* __restrict__ out) {
    const int i = blockIdx.x;
    const int t = threadIdx.x;
    const int tid = token_ids[i];
    const bool valid = edge_len[i] > 1;
    float v = valid ? node_out[(size_t)i * D256 + t] : q[(size_t)tid * D256 + t];
    out[(size_t)tid * D256 + t] = v;
}

__global__ __launch_bounds__(256) void k_fix_pair(const float* __restrict__ node_out,
                                                  const float* __restrict__ path_feat,
                                                  const float* __restrict__ W_pagg,
                                                  const int* __restrict__ token_ids,
                                                  const int* __restrict__ edge_len,
                                                  const int* __restrict__ pair,
                                                  float* __restrict__ out) {
    const int i = blockIdx.x;
    const int t = threadIdx.x;
    const int src = pair[1], tar = pair[0];
    const int tid = token_ids[i];
    if (tid != src && tid != tar) return;
    if (edge_len[i] <= 1) return;
    __shared__ float no[256], pf[256];
    no[t] = node_out[(size_t)i * D256 + t];
    pf[t] = path_feat[t];
    __syncthreads();
    const bool isSrc = (tid == src);
    const float* Wa = W_pagg + (size_t)t * 512 + (isSrc ? 0 : 256);
    const float* Wb = W_pagg + (size_t)t * 512 + (isSrc ? 256 : 0);
    float acc = 0.f;
    for (int ii = 0; ii < 256; ++ii) acc += no[ii] * Wa[ii] + pf[ii] * Wb[ii];
    out[(size_t)tid * D256 + t] = acc;
}

// ---------------------------------------------------------------------------
// Host launcher
// ---------------------------------------------------------------------------
extern "C" void kernel_launch(void* const* d_in, const int* in_sizes, int n_in,
                              void* d_out, int out_size, void* d_ws, size_t ws_size,
                              hipStream_t stream) {
    const float* q       = (const float*)d_in[0];
    const float* k       = (const float*)d_in[1];
    const float* v       = (const float*)d_in[2];
    const float* path    = (const float*)d_in[3];
    const float* W_qpair = (const float*)d_in[4];
    const float* W_q     = (const float*)d_in[5];
    const float* W_k     = (const float*)d_in[6];
    const float* W_v     = (const float*)d_in[7];
    const float* W_pk    = (const float*)d_in[8];
    const float* W_pv    = (const float*)d_in[9];
    const float* W_out   = (const float*)d_in[10];
    const float* b_out   = (const float*)d_in[11];
    const float* W_pagg  = (const float*)d_in[12];
    const int* token_ids = (const int*)d_in[13];
    const int* edge_len  = (const int*)d_in[14];
    const int* seg_ids   = (const int*)d_in[15];
    const int* pair      = (const int*)d_in[16];

    const int S = in_sizes[0] / D256;
    const int E = in_sizes[1] / D256;
    const int P = in_sizes[3] / D256;
    const float scale = 1.0f / 16.0f;  // 1/sqrt(256)

    float* out = (float*)d_out;
    float* wsf = (float*)d_ws;

    size_t o = 0;
    auto alloc = [&](size_t n) { float* p = wsf + o; o += (n + 15) & ~(size_t)15; return p; };
    float* qp        = alloc((size_t)S * D256);
    float* kp        = alloc((size_t)E * D256);
    float* vp        = alloc((size_t)E * D256);
    float* logits    = alloc((size_t)E);
    float* node_vals = alloc((size_t)S * D256);
    float* node_out  = alloc((size_t)S * D256);
    float* Wtp       = alloc((size_t)D256 * D256);
    float* pk        = alloc((size_t)P * D256);
    float* pv        = alloc((size_t)P * D256);
    float* pair_q    = alloc(256);
    float* p_attn    = alloc((size_t)P);
    float* path_feat = alloc(256);
    int*   off       = (int*)alloc((size_t)S + 16);

    const dim3 tb(32, 8);
    const dim3 tg(8, 8);
    const size_t GEMM_LDS = 2 * 16384 * sizeof(float);  // 128 KB double buffer
    (void)hipFuncSetAttribute((const void*)k_gemm_xwt,
                              hipFuncAttributeMaxDynamicSharedMemorySize,
                              (int)GEMM_LDS);

    // ---- path branch (tiny) ----
    k_transpose_pack256<<<tg, tb, 0, stream>>>(W_pk, Wtp);
    k_gemm_xwt<<<(P + 127) / 128, 256, GEMM_LDS, stream>>>(path, Wtp, nullptr, pk, P);
    k_transpose_pack256<<<tg, tb, 0, stream>>>(W_pv, Wtp);
    k_gemm_xwt<<<(P + 127) / 128, 256, GEMM_LDS, stream>>>(path, Wtp, nullptr, pv, P);
    k_pair_q<<<1, 256, 0, stream>>>(q, W_qpair, pair, pair_q);
    k_path_softmax<<<1, 256, 0, stream>>>(pk, pair_q, p_attn, P, scale);
    k_path_feat<<<1, 256, 0, stream>>>(p_attn, pv, W_out, b_out, path_feat, P);

    // ---- big projections (WMMA GEMMs, async-LDS staged weights) ----
    k_transpose_pack256<<<tg, tb, 0, stream>>>(W_q, Wtp);
    k_gemm_xwt<<<(S + 127) / 128, 256, GEMM_LDS, stream>>>(q, Wtp, nullptr, qp, S);
    k_transpose_pack256<<<tg, tb, 0, stream>>>(W_k, Wtp);
    k_gemm_xwt<<<(E + 127) / 128, 256, GEMM_LDS, stream>>>(k, Wtp, nullptr, kp, E);
    k_transpose_pack256<<<tg, tb, 0, stream>>>(W_v, Wtp);
    k_gemm_xwt<<<(E + 127) / 128, 256, GEMM_LDS, stream>>>(v, Wtp, nullptr, vp, E);

    // ---- segment attention ----
    k_scan_offsets<<<1, 256, 0, stream>>>(edge_len, off, S);
    k_edge_logits<<<(E + 7) / 8, 256, 0, stream>>>(qp, kp, seg_ids, logits, E, scale);
    k_node_attn<<<S, 256, 0, stream>>>(logits, vp, off, node_vals);

    // ---- output projection ----
    k_transpose_pack256<<<tg, tb, 0, stream>>>(W_out, Wtp);
    k_gemm_xwt<<<(S + 127) / 128, 256, GEMM_LDS, stream>>>(node_vals, Wtp, b_out, node_out, S);

    // ---- assemble output ----
    const int n4 = S * D256 / 4;
    k_copy_q<<<(n4 + 255) / 256, 256, 0, stream>>>(q, out, n4);
    k_combine<<<S, 256, 0, stream>>>(node_out, q, token_ids, edge_len, out);
    k_fix_pair<<<S, 256, 0, stream>>>(node_out, path_feat, W_pagg, token_ids,
                                      edge_len, pair, out);
}